// GSE_84722524880902
// MI455X (gfx1250) — compile-verified
//
#include <hip/hip_runtime.h>

// Problem constants (from the reference): x[1024,200] f32, W[200,3,512] f32,
// cls[1,1,512], pe[201,512] -> out[1024,201,512] f32.
#define B_   1024
#define C_   200
#define L_   3
#define D_   512
#define SEQ_ 201   // C_ + 1

typedef float v2f __attribute__((ext_vector_type(2)));
typedef float v8f __attribute__((ext_vector_type(8)));

// One block (8 waves, wave32) computes a full 16(batch) x 512(dim) slab for one
// channel c. Each wave owns 64 columns = 4 WMMA tiles sharing one A matrix:
// a single V_WMMA_F32_16X16X4_F32 per 16x16 tile (K=3 window zero-padded to 4).
// All lane-dependent values are selected branchlessly so EXEC stays all-ones.
__global__ __launch_bounds__(256) void band_embed_wmma_kernel(
    const float* __restrict__ x,   // [B_, C_]
    const float* __restrict__ W,   // [C_, L_, D_]
    const float* __restrict__ pe,  // [SEQ_, D_]
    float* __restrict__ out)       // [B_, SEQ_, D_]
{
    const int lane = threadIdx.x & 31;
    const int wv   = threadIdx.x >> 5;       // 0..7 waves per block
    const int c     = blockIdx.x % C_;       // channel
    const int bTile = blockIdx.x / C_;       // 64 b-tiles of 16
    const int n0    = wv << 6;               // 64 columns per wave
    const int b0    = bTile << 4;

    const int laneLo = lane & 15;
    const int laneHi = lane >> 4;            // 0: K={0,1}, 1: K={2,3}

    // ---- A tile: 16x4 f32 (M = batch row = lane%16, K = window tap) ----
    // win[b,c,l] = x[b, clamp(c-1+l, 0, C_-1)]; K=3 is zero padding.
    const int cl = (c == 0) ? 0 : (c - 1);
    const int cr = (c == C_ - 1) ? (C_ - 1) : (c + 1);
    const float* xr = x + (size_t)(b0 + laneLo) * C_;
    const float x_sel = xr[laneHi ? cr : cl];   // K=0 (lo half) / K=2 (hi half)
    const float x_mid = xr[c];                  // K=1 (lo half only)
    v2f a;
    a.x = x_sel;
    a.y = laneHi ? 0.0f : x_mid;

    // ---- 4 B tiles (4x16 f32 each) + 4 WMMAs sharing A ----
    const float* Wc = W + (size_t)c * (L_ * D_) + n0 + laneLo;
    const int kb = laneHi ? 2 : 0;           // K row supplying b.x per lane half
    const float* per = pe + (size_t)(c + 1) * D_ + n0 + laneLo;

    v8f acc[4];
    float pev[4];
    #pragma unroll
    for (int t = 0; t < 4; ++t) {
        const float* Wt = Wc + t * 16;
        const float w_sel = Wt[kb * D_];     // K=0 / K=2
        const float w_mid = Wt[1 * D_];      // K=1 (lo half only)
        v2f bm;
        bm.x = w_sel;
        bm.y = laneHi ? 0.0f : w_mid;

        v8f z = {};
        acc[t] = __builtin_amdgcn_wmma_f32_16x16x4_f32(
            /*neg_a=*/false, a, /*neg_b=*/false, bm,
            /*c_mod=*/(short)0, z, /*reuse_a=*/false, /*reuse_b=*/false);

        pev[t] = per[t * 16];
    }

    // ---- epilogue + stores ----
    // C/D layout: VGPR r -> rows M=r (lanes 0-15) and M=r+8 (lanes 16-31),
    // N = lane%16. One base address; all 32 stores use immediate offsets
    // (max 2,881,728 B < 2^23).
    float* obase = out + ((size_t)(b0 + laneHi * 8) * SEQ_ + (c + 1)) * D_ + n0 + laneLo;
    #pragma unroll
    for (int t = 0; t < 4; ++t) {
        #pragma unroll
        for (int r = 0; r < 8; ++r) {
            obase[t * 16 + (size_t)r * (SEQ_ * D_)] = acc[t][r] + pev[t];
        }
    }
}

// out[b, 0, :] = cls[:] + pe[0, :]  (broadcast over batch), float4 stores.
__global__ __launch_bounds__(256) void cls_pe_kernel(
    const float* __restrict__ cls, const float* __restrict__ pe,
    float* __restrict__ out)
{
    const int tid = blockIdx.x * 256 + threadIdx.x;
    const int n4 = tid & (D_ / 4 - 1);       // 128 float4 per row
    const int b  = tid >> 7;
    const float4 cv = ((const float4*)cls)[n4];
    const float4 pv = ((const float4*)pe)[n4];
    float4 v;
    v.x = cv.x + pv.x; v.y = cv.y + pv.y;
    v.z = cv.z + pv.z; v.w = cv.w + pv.w;
    ((float4*)out)[(size_t)b * (SEQ_ * D_ / 4) + n4] = v;
}

extern "C" void kernel_launch(void* const* d_in, const int* in_sizes, int n_in,
                              void* d_out, int out_size, void* d_ws, size_t ws_size,
                              hipStream_t stream) {
    (void)in_sizes; (void)n_in; (void)out_size; (void)d_ws; (void)ws_size;
    const float* x   = (const float*)d_in[0];   // [1024, 200]
    const float* W   = (const float*)d_in[1];   // [200, 3, 512]
    const float* cls = (const float*)d_in[2];   // [1, 1, 512]
    const float* pe  = (const float*)d_in[3];   // [201, 512]
    float* out = (float*)d_out;                 // [1024, 201, 512]

    // One block per (b-tile, channel): 64 * 200 = 12800 blocks of 256 threads.
    band_embed_wmma_kernel<<<(B_ / 16) * C_, 256, 0, stream>>>(x, W, pe, out);

    // cls row: B_*D_/4 = 131072 float4 -> 512 blocks of 256.
    cls_pe_kernel<<<(B_ * D_ / 4) / 256, 256, 0, stream>>>(cls, pe, out);
}